// MultiHeadModulator_76338748719651
// MI455X (gfx1250) — compile-verified
//
#include <hip/hip_runtime.h>
#include <hip/hip_bf16.h>
#include <math.h>

// MultiHeadModulator, algebraically reduced:
//   scores[l,h] = (z_past[l,:].U[h,:] + c[h] + R[idx(l),h]) / sqrt(32)
//   w = softmax_L(scores);  s[h,:] = sum_l w[l,h] * z_past[l,:]
//   out_pre[j] = s[h(j)].Wv[j,:] + bv[j];  out = out_pre @ Wo^T + bo
// Two streaming passes over z_past (128 MB each; fits in 192 MB L2), both
// implemented with V_WMMA_F32_16X16X4_F32 (heads padded 8 -> 16).

typedef float f32x2 __attribute__((ext_vector_type(2)));
typedef float f32x8 __attribute__((ext_vector_type(8)));

#define D2 512           // 2*dim
#define NHEADS 8
#define HREALS 64        // reals per head
#define RELN 129         // 2*REL_MAX+1

// ---------------- Kernel A: q_proj, U, c, R (single block) ----------------
__global__ void prep_kernel(const float* __restrict__ z_curr,
                            const float* __restrict__ Wq, const float* __restrict__ bq,
                            const float* __restrict__ Wk, const float* __restrict__ bk,
                            const float* __restrict__ rel,
                            float* __restrict__ qp, float* __restrict__ U,
                            float* __restrict__ cbias, float* __restrict__ R) {
  const int t = threadIdx.x;  // 256 threads
  for (int j = t; j < D2; j += 256) {
    const float* wr = Wq + (size_t)j * D2;
    float acc = bq[j];
    for (int i = 0; i < D2; ++i) acc += wr[i] * z_curr[i];
    qp[j] = acc;
  }
  __syncthreads();
  // U[h][i] = sum_{j in head h} qp[j] * Wk[j][i]   (rows 8..15 zero-padded)
  for (int i = t; i < D2; i += 256) {
    for (int h = 0; h < 16; ++h) {
      float acc = 0.f;
      if (h < NHEADS) {
        const int j0 = h * HREALS;
        for (int j = 0; j < HREALS; ++j) acc += qp[j0 + j] * Wk[(size_t)(j0 + j) * D2 + i];
      }
      U[h * D2 + i] = acc;
    }
  }
  if (t < 16) {
    float acc = 0.f;
    if (t < NHEADS) {
      const int j0 = t * HREALS;
      for (int j = 0; j < HREALS; ++j) acc += qp[j0 + j] * bk[j0 + j];
    }
    cbias[t] = acc;
  }
  // R[t][h] = sum_{j in head h} qp[j] * rel[t][j]
  for (int e = t; e < RELN * 16; e += 256) {
    const int h = e & 15, rrow = e >> 4;
    float acc = 0.f;
    if (h < NHEADS) {
      const int j0 = h * HREALS;
      const float* rr = rel + (size_t)rrow * D2;
      for (int j = 0; j < HREALS; ++j) acc += qp[j0 + j] * rr[j0 + j];
    }
    R[rrow * 16 + h] = acc;
  }
}

// ---------- Kernel B: scores via WMMA, streaming z_past (128 rows/block) ----------
__global__ void scores_kernel(const float* __restrict__ zp, const float* __restrict__ U,
                              const float* __restrict__ cbias, const float* __restrict__ R,
                              const int* __restrict__ curr_pos_p, int L,
                              float* __restrict__ scores, float* __restrict__ blockmax) {
  __shared__ float wmax[8 * 8];  // [wave][head]
  const int tid = threadIdx.x;
  const int wave = tid >> 5, lane = tid & 31;
  const int m = lane & 15;            // row-in-tile for A, head/col for B
  const int khalf = (lane >> 4) * 2;  // K sub-offset per lane half
  const int rowbase = blockIdx.x * 128 + wave * 16;
  const int curr = *curr_pos_p;
  const float scale = 0.17677669529663687f;  // 1/sqrt(32)

  f32x8 c = {};
  const float* arow = zp + (size_t)(rowbase + m) * D2 + khalf;  // A[m][k]
  const float* brow = U + (size_t)m * D2 + khalf;               // B[k][n] = U[n][k]
  for (int k = 0; k < D2; k += 4) {
    if ((k & 127) == 0) __builtin_prefetch(arow + k + 256, 0, 1);  // global_prefetch_b8
    f32x2 a, b;
    a.x = arow[k]; a.y = arow[k + 1];
    b.x = brow[k]; b.y = brow[k + 1];
    c = __builtin_amdgcn_wmma_f32_16x16x4_f32(false, a, false, b, (short)0, c, false, false);
  }

  // C layout: lanes 0-15: M=r, N=lane; lanes 16-31: M=8+r, N=lane-16
  const int h = m;
  const int mofs = (lane < 16) ? 0 : 8;
  const float cb = (h < NHEADS) ? cbias[h] : 0.f;
  float hmax = -3.4e38f;
  for (int r = 0; r < 8; ++r) {
    const int srow = rowbase + mofs + r;
    if (h < NHEADS) {
      int rel_d = curr - L + srow + 64;  // + REL_MAX
      rel_d = rel_d < 0 ? 0 : (rel_d > 128 ? 128 : rel_d);
      const float val = (c[r] + cb + R[rel_d * 16 + h]) * scale;
      scores[(size_t)srow * NHEADS + h] = val;
      hmax = fmaxf(hmax, val);
    }
  }
  hmax = fmaxf(hmax, __shfl_xor(hmax, 16, 32));  // combine the two M-halves
  if (lane < NHEADS) wmax[wave * 8 + h] = hmax;
  __syncthreads();
  if (tid < NHEADS) {
    float mx = wmax[tid];
    for (int w = 1; w < 8; ++w) mx = fmaxf(mx, wmax[w * 8 + tid]);
    blockmax[(size_t)blockIdx.x * NHEADS + tid] = mx;
  }
}

// ---------------- Kernel C: global max (fixed order, deterministic) ----------------
__global__ void gmax_kernel(const float* __restrict__ blockmax, int NB, float* __restrict__ gmax) {
  const int h = threadIdx.x;
  if (h < NHEADS) {
    float mx = blockmax[h];
    for (int b = 1; b < NB; ++b) mx = fmaxf(mx, blockmax[(size_t)b * NHEADS + h]);
    gmax[h] = mx;
  }
}

// ---------------- Kernel D: exp in-place + per-block sums ----------------
__global__ void exp_kernel(float* __restrict__ scores, const float* __restrict__ gmax,
                           float* __restrict__ blocksum) {
  __shared__ float acc[256 * 8];
  const int t = threadIdx.x;
  const size_t base = (size_t)blockIdx.x * 1024;  // 128 rows * 8 heads
  float g[8], la[8];
  for (int h = 0; h < 8; ++h) { g[h] = gmax[h]; la[h] = 0.f; }
  for (int q = 0; q < 4; ++q) {
    const size_t gi = base + (size_t)q * 256 + t;
    const int h = (int)(gi & 7);
    const float e = __expf(scores[gi] - g[h]);
    scores[gi] = e;
    la[h] += e;
  }
  for (int h = 0; h < 8; ++h) acc[t * 8 + h] = la[h];
  __syncthreads();
  for (int s = 128; s > 0; s >>= 1) {
    if (t < s)
      for (int h = 0; h < 8; ++h) acc[t * 8 + h] += acc[(t + s) * 8 + h];
    __syncthreads();
  }
  if (t < 8) blocksum[(size_t)blockIdx.x * 8 + t] = acc[t];
}

// ---------------- Kernel D2: global sums (fixed order) ----------------
__global__ void gsum_kernel(const float* __restrict__ blocksum, int NB, float* __restrict__ S) {
  const int h = threadIdx.x;
  if (h < NHEADS) {
    float s = 0.f;
    for (int b = 0; b < NB; ++b) s += blocksum[(size_t)b * NHEADS + h];
    S[h] = s;
  }
}

// ------- Kernel E: s partials = e^T @ z_past via WMMA (1 wave per block) -------
// grid = (32 column strips of 16, KB chunks of 512 rows); deterministic partials.
__global__ void wsum_kernel(const float* __restrict__ zp, const float* __restrict__ e,
                            float* __restrict__ pd) {
  const int lane = threadIdx.x & 31;
  const int m = lane & 15;
  const int khalf = (lane >> 4) * 2;
  const int col0 = blockIdx.x * 16;
  const int kb = blockIdx.y;
  const int r0 = kb * 512;

  f32x8 c = {};
  const float* bcol = zp + (size_t)r0 * D2 + col0 + m;                         // B[k][n]
  const float* acol = (m < NHEADS) ? (e + (size_t)r0 * NHEADS + m) : nullptr;  // A[m][k]
  for (int k = 0; k < 512; k += 4) {
    const int kk = k + khalf;
    f32x2 a, b;
    if (m < NHEADS) { a.x = acol[(size_t)kk * NHEADS]; a.y = acol[(size_t)(kk + 1) * NHEADS]; }
    else            { a.x = 0.f; a.y = 0.f; }
    b.x = bcol[(size_t)kk * D2];
    b.y = bcol[(size_t)(kk + 1) * D2];
    c = __builtin_amdgcn_wmma_f32_16x16x4_f32(false, a, false, b, (short)0, c, false, false);
  }
  const int mofs = (lane < 16) ? 0 : 8;
  float* out = pd + (size_t)kb * (16 * D2);
  for (int r = 0; r < 8; ++r) out[(size_t)(mofs + r) * D2 + col0 + m] = c[r];
}

// ---------------- Kernel F: reduce K-chunk partials (fixed order) ----------------
__global__ void reduce_pd_kernel(const float* __restrict__ pd, int KB, float* __restrict__ sacc) {
  const int idx = blockIdx.x * 256 + threadIdx.x;  // 16*512 entries
  float a = 0.f;
  for (int kb = 0; kb < KB; ++kb) a += pd[(size_t)kb * (16 * D2) + idx];
  sacc[idx] = a;
}

// ---------------- Kernel G: output projections (single block) ----------------
__global__ void out_kernel(const float* __restrict__ sacc, const float* __restrict__ S,
                           const float* __restrict__ Wv, const float* __restrict__ bv,
                           const float* __restrict__ Wo, const float* __restrict__ bo,
                           float* __restrict__ dout) {
  __shared__ float op[D2];
  const int t = threadIdx.x;
  for (int j = t; j < D2; j += 256) {
    const int h = j >> 6;  // 64 reals per head
    const float* s = sacc + (size_t)h * D2;
    const float* w = Wv + (size_t)j * D2;
    float acc = 0.f;
    for (int i = 0; i < D2; ++i) acc += s[i] * w[i];
    op[j] = acc / S[h] + bv[j];
  }
  __syncthreads();
  for (int r = t; r < D2; r += 256) {
    const float* w = Wo + (size_t)r * D2;
    float acc = bo[r];
    for (int j = 0; j < D2; ++j) acc += op[j] * w[j];
    dout[r] = acc;
  }
}

extern "C" void kernel_launch(void* const* d_in, const int* in_sizes, int n_in,
                              void* d_out, int out_size, void* d_ws, size_t ws_size,
                              hipStream_t stream) {
  const int*   curr_pos = (const int*)d_in[0];
  const float* z_curr   = (const float*)d_in[1];
  const float* z_past   = (const float*)d_in[2];
  const float* Wq = (const float*)d_in[3];  const float* bq = (const float*)d_in[4];
  const float* Wk = (const float*)d_in[5];  const float* bk = (const float*)d_in[6];
  const float* Wv = (const float*)d_in[7];  const float* bv = (const float*)d_in[8];
  const float* Wo = (const float*)d_in[9];  const float* bo = (const float*)d_in[10];
  const float* rel = (const float*)d_in[11];
  float* out = (float*)d_out;

  const int L  = in_sizes[2] / D2;   // past_len (65536)
  const int NB = L / 128;            // score blocks (512)
  const int KB = L / 512;            // K-chunks for weighted sum (128)

  float* ws = (float*)d_ws;
  size_t o = 0;
  float* qp    = ws + o; o += D2;
  float* U     = ws + o; o += 16 * D2;
  float* cbias = ws + o; o += 16;
  float* R     = ws + o; o += RELN * 16;
  float* gmax  = ws + o; o += 8;
  float* S     = ws + o; o += 8;
  float* sacc  = ws + o; o += 16 * D2;
  float* bmax  = ws + o; o += (size_t)NB * 8;
  float* bsum  = ws + o; o += (size_t)NB * 8;
  float* sc    = ws + o; o += (size_t)L * 8;      // scores -> exp in place
  float* pd    = ws + o; o += (size_t)KB * 16 * D2;

  prep_kernel<<<1, 256, 0, stream>>>(z_curr, Wq, bq, Wk, bk, rel, qp, U, cbias, R);
  scores_kernel<<<NB, 256, 0, stream>>>(z_past, U, cbias, R, curr_pos, L, sc, bmax);
  gmax_kernel<<<1, 32, 0, stream>>>(bmax, NB, gmax);
  exp_kernel<<<NB, 256, 0, stream>>>(sc, gmax, bsum);
  gsum_kernel<<<1, 32, 0, stream>>>(bsum, NB, S);
  wsum_kernel<<<dim3(D2 / 16, KB), 32, 0, stream>>>(z_past, sc, pd);
  reduce_pd_kernel<<<(16 * D2) / 256, 256, 0, stream>>>(pd, KB, sacc);
  out_kernel<<<1, 256, 0, stream>>>(sacc, S, Wv, bv, Wo, bo, out);
}